// MyDGSR_8452495638540
// MI455X (gfx1250) — compile-verified
//
#include <hip/hip_runtime.h>

#define NL   50      // neighbors per node
#define DD   64      // feature dim
#define ROWS 16      // destination nodes per block
#define NT   256     // threads per block (8 wave32)

typedef __attribute__((ext_vector_type(2))) float v2f;
typedef __attribute__((ext_vector_type(8))) float v8f;
typedef int v4i __attribute__((vector_size(16)));

#if __has_builtin(__builtin_amdgcn_global_load_async_to_lds_b128) && \
    __has_builtin(__builtin_amdgcn_s_wait_asynccnt)
#define USE_ASYNC_LDS 1
#else
#define USE_ASYNC_LDS 0
#endif

#if USE_ASYNC_LDS
typedef __attribute__((address_space(1))) v4i* gv4i_p;   // global int4*
typedef __attribute__((address_space(3))) v4i* lv4i_p;   // LDS int4*
__device__ __forceinline__ void async_b128(const void* g, void* l) {
  __builtin_amdgcn_global_load_async_to_lds_b128(
      (gv4i_p)(v4i*)const_cast<void*>(g), (lv4i_p)(v4i*)l, 0, 0);
}
#endif

struct __align__(16) Smem {
  float tenc[NL][DD];     // 12800 B  time encoding table
  float dstf[ROWS][DD];   //  4096 B  destination features
  float h[ROWS][DD];      //  4096 B  attention output
  float act[ROWS][DD];    //  4096 B  MLP hidden activations
  float ep[ROWS][NL];     //  3200 B  logits -> alpha
  float rowsum[ROWS];     //    64 B  rank-prior softmax denom
  int   times[ROWS][NL];  //  3200 B
  int   idx[ROWS][NL];    //  3200 B
  int   srt[ROWS][NL];    //  3200 B  srt[rank] = neighbor slot
  int   reord[ROWS][NL];  //  3200 B  L-1-rank per slot
};

__global__ __launch_bounds__(NT) void orgat_side_kernel(
    const float* __restrict__ dst_h, const float* __restrict__ src_h,
    const float* __restrict__ tenc,
    const float* __restrict__ W1, const float* __restrict__ b1,
    const float* __restrict__ W2, const float* __restrict__ b2,
    const int* __restrict__ nbr_idx, const int* __restrict__ nbr_time,
    float* __restrict__ out)
{
  __shared__ Smem s;
  const int tid  = threadIdx.x;
  const int base = blockIdx.x * ROWS;   // 50000 % 16 == 0, no bounds checks

  // ---------- stage 0: bulk global -> LDS (async path when available) ----------
#if USE_ASYNC_LDS
  {
    const float4* g0 = (const float4*)tenc;
    float4* d0 = (float4*)&s.tenc[0][0];
    for (int i = tid; i < NL * DD / 4; i += NT) async_b128(g0 + i, d0 + i);

    const float4* g1 = (const float4*)(dst_h + (size_t)base * DD);
    float4* d1 = (float4*)&s.dstf[0][0];
    for (int i = tid; i < ROWS * DD / 4; i += NT) async_b128(g1 + i, d1 + i);

    const int4* g2 = (const int4*)(nbr_time + (size_t)base * NL); // blk*3200B, 16B aligned
    int4* d2 = (int4*)&s.times[0][0];
    for (int i = tid; i < ROWS * NL / 4; i += NT) async_b128(g2 + i, d2 + i);

    const int4* g3 = (const int4*)(nbr_idx + (size_t)base * NL);
    int4* d3 = (int4*)&s.idx[0][0];
    for (int i = tid; i < ROWS * NL / 4; i += NT) async_b128(g3 + i, d3 + i);

    __builtin_amdgcn_s_wait_asynccnt(0);
  }
#else
  {
    const float4* g = (const float4*)tenc;
    float4* d = (float4*)&s.tenc[0][0];
    for (int i = tid; i < NL * DD / 4; i += NT) d[i] = g[i];
  }
  {
    const float4* g = (const float4*)(dst_h + (size_t)base * DD);
    float4* d = (float4*)&s.dstf[0][0];
    for (int i = tid; i < ROWS * DD / 4; i += NT) d[i] = g[i];
  }
  {
    const int4* gt = (const int4*)(nbr_time + (size_t)base * NL);
    int4* dt = (int4*)&s.times[0][0];
    for (int i = tid; i < ROWS * NL / 4; i += NT) dt[i] = gt[i];
    const int4* gi = (const int4*)(nbr_idx + (size_t)base * NL);
    int4* di = (int4*)&s.idx[0][0];
    for (int i = tid; i < ROWS * NL / 4; i += NT) di[i] = gi[i];
  }
#endif
  __syncthreads();

  // ---------- stage 1: stable rank per row (matches stable argsort) ----------
  for (int p = tid; p < ROWS * NL; p += NT) {
    const int r = p / NL, i = p - r * NL;
    const int ti = s.times[r][i];
    int rank = 0;
    for (int j = 0; j < NL; ++j) {
      const int tj = s.times[r][j];
      rank += (tj < ti) | ((tj == ti) & (j < i));
    }
    s.reord[r][i]  = NL - 1 - rank;   // re_order = L - rank - 1
    s.srt[r][rank] = i;               // srt = argsort(times)
  }
  __syncthreads();

  // rank-prior softmax denominator: max(srt) is always NL-1
  if (tid < ROWS) {
    float sum = 0.f;
    for (int l = 0; l < NL; ++l)
      sum += __expf((float)s.srt[tid][l] - (float)(NL - 1));
    s.rowsum[tid] = sum;
  }
  __syncthreads();

  // ---------- stage 2: logits e*p  (gather pass 1, L2-resident table) ----------
  for (int p = tid; p < ROWS * NL; p += NT) {
    const int r = p / NL, l = p - r * NL;
    const float4* qrow = (const float4*)(src_h + (size_t)s.idx[r][l] * DD);
    const float4* te   = (const float4*)&s.tenc[s.reord[r][l]][0];
    const float4* dr   = (const float4*)&s.dstf[r][0];
    float acc = 0.f;
    #pragma unroll
    for (int d4 = 0; d4 < DD / 4; ++d4) {
      const float4 a = qrow[d4], t = te[d4], w = dr[d4];
      acc = fmaf(a.x + t.x, w.x, acc);
      acc = fmaf(a.y + t.y, w.y, acc);
      acc = fmaf(a.z + t.z, w.z, acc);
      acc = fmaf(a.w + t.w, w.w, acc);
    }
    const float pr = __expf((float)s.srt[r][l] - (float)(NL - 1)) / s.rowsum[r];
    s.ep[r][l] = acc * 6.25f * pr;    // scale = L/sqrt(D) = 50/8
  }
  __syncthreads();

  // ---------- attention softmax over L per row ----------
  if (tid < ROWS) {
    float m = -3.4e38f;
    for (int l = 0; l < NL; ++l) m = fmaxf(m, s.ep[tid][l]);
    float sum = 0.f;
    for (int l = 0; l < NL; ++l) {
      const float v = __expf(s.ep[tid][l] - m);
      s.ep[tid][l] = v; sum += v;
    }
    const float inv = 1.0f / sum;
    for (int l = 0; l < NL; ++l) s.ep[tid][l] *= inv;
  }
  __syncthreads();

  // ---------- stage 3: h = alpha @ nbr_h  (gather pass 2) ----------
  {
    const int r  = tid >> 4;           // 16 threads per row
    const int dq = (tid & 15) << 2;    // 4 floats each
    float4 acc = make_float4(0.f, 0.f, 0.f, 0.f);
    for (int l = 0; l < NL; ++l) {
      if (l + 4 < NL)
        __builtin_prefetch(src_h + (size_t)s.idx[r][l + 4] * DD + dq, 0, 0);
      const float  a = s.ep[r][l];
      const float4 v = *(const float4*)(src_h + (size_t)s.idx[r][l] * DD + dq);
      acc.x = fmaf(a, v.x, acc.x);
      acc.y = fmaf(a, v.y, acc.y);
      acc.z = fmaf(a, v.z, acc.z);
      acc.w = fmaf(a, v.w, acc.w);
    }
    *(float4*)&s.h[r][dq] = acc;
  }
  __syncthreads();

  // ---------- stage 4: MLP with V_WMMA_F32_16X16X4_F32 ----------
  // A 16x4 f32: lane l -> M = l&15, K = 2*(l>>4)+v.  B 4x16 mirrored.
  // C/D 16x16 f32: M = v + 8*(l>>4), N = l&15.
  const int w    = tid >> 5;       // wave id (0..7); waves 0..3 do the GEMMs
  const int lane = tid & 31;
  const int ln   = lane & 15;
  const int kh   = lane >> 4;

  if (w < 4) {                     // layer 1: act = relu(h @ W1 + b1)
    const int c = w << 4;
    v8f acc = {};
    #pragma unroll
    for (int k0 = 0; k0 < DD; k0 += 4) {
      const int k = k0 + (kh << 1);
      v2f a, b;
      a.x = s.h[ln][k];
      a.y = s.h[ln][k + 1];
      b.x = W1[(size_t)k * DD + c + ln];
      b.y = W1[(size_t)(k + 1) * DD + c + ln];
      acc = __builtin_amdgcn_wmma_f32_16x16x4_f32(
          false, a, false, b, (short)0, acc, false, false);
    }
    const float bb = b1[c + ln];
    #pragma unroll
    for (int v = 0; v < 8; ++v)
      s.act[v + (kh << 3)][c + ln] = fmaxf(acc[v] + bb, 0.0f);
  }
  __syncthreads();

  if (w < 4) {                     // layer 2: out = act @ W2 + b2
    const int c = w << 4;
    v8f acc = {};
    #pragma unroll
    for (int k0 = 0; k0 < DD; k0 += 4) {
      const int k = k0 + (kh << 1);
      v2f a, b;
      a.x = s.act[ln][k];
      a.y = s.act[ln][k + 1];
      b.x = W2[(size_t)k * DD + c + ln];
      b.y = W2[(size_t)(k + 1) * DD + c + ln];
      acc = __builtin_amdgcn_wmma_f32_16x16x4_f32(
          false, a, false, b, (short)0, acc, false, false);
    }
    const float bb = b2[c + ln];
    #pragma unroll
    for (int v = 0; v < 8; ++v)
      out[(size_t)(base + v + (kh << 3)) * DD + c + ln] = acc[v] + bb;
  }
}

extern "C" void kernel_launch(void* const* d_in, const int* in_sizes, int n_in,
                              void* d_out, int out_size, void* d_ws, size_t ws_size,
                              hipStream_t stream) {
  const float* user_h = (const float*)d_in[0];
  const float* item_h = (const float*)d_in[1];
  const float* u_enc  = (const float*)d_in[2];
  const float* i_enc  = (const float*)d_in[3];
  const float* Wu1 = (const float*)d_in[4];
  const float* bu1 = (const float*)d_in[5];
  const float* Wu2 = (const float*)d_in[6];
  const float* bu2 = (const float*)d_in[7];
  const float* Wi1 = (const float*)d_in[8];
  const float* bi1 = (const float*)d_in[9];
  const float* Wi2 = (const float*)d_in[10];
  const float* bi2 = (const float*)d_in[11];
  const int* u_nbr_items = (const int*)d_in[12];
  const int* u_nbr_time  = (const int*)d_in[13];
  const int* i_nbr_users = (const int*)d_in[14];
  const int* i_nbr_time  = (const int*)d_in[15];

  float* out = (float*)d_out;
  const int N_USERS = 50000, N_ITEMS = 50000;

  // side 'by': item -> user
  orgat_side_kernel<<<dim3(N_USERS / ROWS), NT, 0, stream>>>(
      user_h, item_h, u_enc, Wu1, bu1, Wu2, bu2,
      u_nbr_items, u_nbr_time, out);

  // side 'pby': user -> item
  orgat_side_kernel<<<dim3(N_ITEMS / ROWS), NT, 0, stream>>>(
      item_h, user_h, i_enc, Wi1, bi1, Wi2, bi2,
      i_nbr_users, i_nbr_time, out + (size_t)N_USERS * DD);
}